// Router_30537217474765
// MI455X (gfx1250) — compile-verified
//
#include <hip/hip_runtime.h>
#include <hip/hip_bf16.h>
#include <math.h>

// ---------------------------------------------------------------------------
// MoE router (gate) for MI455X / gfx1250, wave32 + V_WMMA_F32_16X16X4_F32.
//   x:[N,2048] f32, W:[64,2048] f32
//   out = [topk_weight (N*2 f32) | topk_idx as f32 (N*2) | aux_loss (1)]
// HBM-bound on streaming x (134 MB ~ 5.8us at 23.3 TB/s); fp32 WMMA keeps
// compute at a comparable rate without precision loss vs the f32 reference.
// ---------------------------------------------------------------------------

typedef float v2f __attribute__((ext_vector_type(2)));
typedef float v8f __attribute__((ext_vector_type(8)));

#define HD        2048   // d_model
#define NE        64     // experts
#define KC        64     // K chunk staged in LDS
#define TPB       128    // tokens per block (8 waves * 16)
#define XS_STRIDE 68     // 128-row x tile stride (float4-aligned, bank-spread)
#define WS_STRIDE 68     // 64-row W tile stride
#define LG_STRIDE 65     // logits stride (conflict-free row scans)
#define ALPHA     0.01f
#define TOPK      2

__global__ void router_init_ws(float* g_sums, unsigned* g_cnt) {
    int i = threadIdx.x;
    if (i < NE) { g_sums[i] = 0.0f; g_cnt[i] = 0u; }
}

__global__ __launch_bounds__(256) void router_main(
    const float* __restrict__ x, const float* __restrict__ W,
    float* __restrict__ out, float* __restrict__ g_sums,
    unsigned* __restrict__ g_cnt, int N)
{
    __shared__ float xs[TPB * XS_STRIDE];        // 34816 B (also reused for logits)
    __shared__ float wsm[NE * WS_STRIDE];        // 17408 B
    __shared__ float sm_sums[NE];
    __shared__ unsigned sm_cnt[NE];

    const int tid  = threadIdx.x;
    const int lane = tid & 31;
    const int wv   = tid >> 5;
    const int tokenBase = blockIdx.x * TPB;

    if (tid < NE) { sm_sums[tid] = 0.0f; sm_cnt[tid] = 0u; }

    v8f acc[4];
    #pragma unroll
    for (int t = 0; t < 4; ++t) acc[t] = (v8f)0.0f;

    const int arow  = (wv << 4) + (lane & 15);   // token row for A frags
    const int khalf = (lane >> 4) << 1;          // 0 (lanes 0-15) / 2 (lanes 16-31)

    for (int k0 = 0; k0 < HD; k0 += KC) {
        __syncthreads();   // previous chunk fully consumed before restaging

        // Stage x tile: 128 rows x 64 cols, 8 float4 per thread, coalesced.
        #pragma unroll
        for (int i = 0; i < 8; ++i) {
            int linear = tid + i * 256;
            int row = linear >> 4;
            int c4  = (linear & 15) << 2;
            float4 v = *(const float4*)(x + (size_t)(tokenBase + row) * HD + k0 + c4);
            *(float4*)&xs[row * XS_STRIDE + c4] = v;
        }
        // Stage W chunk: 64 rows x 64 cols, 4 float4 per thread.
        #pragma unroll
        for (int i = 0; i < 4; ++i) {
            int linear = tid + i * 256;
            int row = linear >> 4;
            int c4  = (linear & 15) << 2;
            *(float4*)&wsm[row * WS_STRIDE + c4] =
                *(const float4*)(W + (size_t)row * HD + k0 + c4);
        }
        // Prefetch next x chunk into the cache hierarchy (global_prefetch_b8).
        if (k0 + KC < HD) {
            __builtin_prefetch(x + (size_t)(tokenBase + (tid >> 1)) * HD
                                 + (k0 + KC) + ((tid & 1) << 5), 0, 3);
        }
        __syncthreads();

        // 16 K-steps of V_WMMA_F32_16X16X4_F32 over 4 expert tiles.
        #pragma unroll
        for (int kk = 0; kk < KC / 4; ++kk) {
            int col = (kk << 2) + khalf;
            v2f a = *(const v2f*)&xs[arow * XS_STRIDE + col];
            #pragma unroll
            for (int t = 0; t < 4; ++t) {
                v2f b = *(const v2f*)&wsm[((t << 4) + (lane & 15)) * WS_STRIDE + col];
                acc[t] = __builtin_amdgcn_wmma_f32_16x16x4_f32(
                    /*neg_a=*/false, a, /*neg_b=*/false, b,
                    /*c_mod=*/(short)0, acc[t],
                    /*reuse_a=*/false, /*reuse_b=*/false);
            }
        }
    }
    __syncthreads();

    // Dump logits to LDS (reuse xs): D layout -> M = r + 8*(lane>>4), Nsub = lane&15.
    float* lg = xs;
    {
        int e   = lane & 15;
        int mhi = (lane >> 4) << 3;
        #pragma unroll
        for (int t = 0; t < 4; ++t)
            #pragma unroll
            for (int r = 0; r < 8; ++r)
                lg[((wv << 4) + mhi + r) * LG_STRIDE + (t << 4) + e] = acc[t][r];
    }
    __syncthreads();

    // Softmax + top-2 + aux accumulation: lanes 0-15 each own one token.
    if (lane < 16) {
        int lt  = (wv << 4) + lane;
        int tok = tokenBase + lt;
        float* row = &lg[lt * LG_STRIDE];

        float m  = -3.402823466e38f;
        float v0 = -3.402823466e38f, v1 = -3.402823466e38f;
        int   i0 = 0, i1 = 0;
        for (int e = 0; e < NE; ++e) {
            float l = row[e];
            m = fmaxf(m, l);
            if (l > v0)      { v1 = v0; i1 = i0; v0 = l; i0 = e; }
            else if (l > v1) { v1 = l; i1 = e; }
        }
        float sum = 0.0f;
        for (int e = 0; e < NE; ++e) {
            float s = __expf(row[e] - m);
            row[e] = s;
            sum += s;
        }
        float inv = 1.0f / sum;
        float s0 = __expf(v0 - m) * inv;
        float s1 = __expf(v1 - m) * inv;
        float wn = 1.0f / (s0 + s1 + 1e-9f);

        out[(size_t)tok * 2 + 0] = s0 * wn;
        out[(size_t)tok * 2 + 1] = s1 * wn;
        out[(size_t)N * 2 + (size_t)tok * 2 + 0] = (float)i0;
        out[(size_t)N * 2 + (size_t)tok * 2 + 1] = (float)i1;

        atomicAdd(&sm_cnt[i0], 1u);
        atomicAdd(&sm_cnt[i1], 1u);
        for (int e = 0; e < NE; ++e)
            atomicAdd(&sm_sums[e], row[e] * inv);   // softmax score sums (Pi numerator)
    }
    __syncthreads();

    if (tid < NE) {
        atomicAdd(&g_sums[tid], sm_sums[tid]);
        atomicAdd(&g_cnt[tid], sm_cnt[tid]);
    }
}

__global__ void router_aux(const float* __restrict__ g_sums,
                           const unsigned* __restrict__ g_cnt,
                           float* __restrict__ out, int N)
{
    __shared__ float red[NE];
    int e = threadIdx.x;
    red[e] = g_sums[e] * (float)g_cnt[e];
    __syncthreads();
    for (int s = NE / 2; s > 0; s >>= 1) {
        if (e < s) red[e] += red[e + s];
        __syncthreads();
    }
    if (e == 0) {
        // aux = ALPHA * sum_e (S[e]/N) * (E * C[e]/(N*K))
        float scale = ALPHA * (float)NE / ((float)N * (float)N * (float)TOPK);
        out[(size_t)N * 4] = scale * red[0];
    }
}

extern "C" void kernel_launch(void* const* d_in, const int* in_sizes, int n_in,
                              void* d_out, int out_size, void* d_ws, size_t ws_size,
                              hipStream_t stream)
{
    const float* x = (const float*)d_in[0];
    const float* W = (const float*)d_in[1];
    float* out = (float*)d_out;
    const int N = in_sizes[0] / HD;    // 16384 tokens

    float*    g_sums = (float*)d_ws;
    unsigned* g_cnt  = (unsigned*)((char*)d_ws + NE * sizeof(float));

    router_init_ws<<<1, 128, 0, stream>>>(g_sums, g_cnt);
    router_main<<<N / TPB, 256, 0, stream>>>(x, W, out, g_sums, g_cnt, N);
    router_aux<<<1, NE, 0, stream>>>(g_sums, g_cnt, out, N);
}